// FrequencyLayer_75479755260642
// MI455X (gfx1250) — compile-verified
//
#include <hip/hip_runtime.h>
#include <hip/hip_bf16.h>
#include <math.h>

// Problem constants (match reference setup_inputs)
#define BATCH 32
#define SEQ   2048
#define HID   1024
#define SCHUNKS 8                  // S split for pass-1 parallelism
#define CHUNK_S (SEQ / SCHUNKS)    // 256 rows per chunk
#define HTILES (HID / 16)          // 64 WMMA column tiles
#define TWO_PI 6.283185307179586476925f

typedef __attribute__((ext_vector_type(2))) float v2f;
typedef __attribute__((ext_vector_type(8))) float v8f;

// ---------------------------------------------------------------------------
// Pass 1: per-(b,h) reductions  D[3,16] += W[3,4] x X[4,16]  via
// V_WMMA_F32_16X16X4_F32, accumulated over a 256-row S chunk per wave.
//
// f32 WMMA register layouts (ISA 7.12.2):
//   A (16x4):  lanes 0-15 -> M=0..15, VGPR0=K0 VGPR1=K1 ; lanes 16-31 -> K2,K3
//   B (4x16):  lanes 0-15 -> N=0..15 rows K0(V0),K1(V1); lanes 16-31 -> K2,K3
//   D (16x16): VGPRj lanes 0-15 -> row M=j  (rows 0..2 are a0/a1/b1 sums)
//
// Weights: row0 = 1/S, row1 = (2/S)cos(theta*s), row2 = (2/S)sin(theta*s).
// cos/sin are advanced by a 4*theta rotation recurrence (no transcendentals
// in the loop -> no divergence, EXEC stays all-ones for WMMA).
// ---------------------------------------------------------------------------
__global__ __launch_bounds__(256) void
freq_pass1_wmma(const float* __restrict__ x, float* __restrict__ partial) {
  const int wave = threadIdx.x >> 5;
  const int lane = threadIdx.x & 31;
  const int gw   = blockIdx.x * 8 + wave;      // 16384 waves total
  const int chunk = gw & (SCHUNKS - 1);
  const int tile  = (gw >> 3) & (HTILES - 1);
  const int b     = gw >> 9;                   // / (SCHUNKS*HTILES)
  const int h0    = tile << 4;
  const int sbase = chunk * CHUNK_S;

  const int n  = lane & 15;                    // B: column within tile
  const int kh = (lane >> 4) << 1;             // 0 (lanes 0-15) or 2 (16-31)
  const int mA = lane & 15;                    // A: row index for this lane

  const float theta  = TWO_PI / (float)SEQ;
  const float w0     = 1.0f / (float)SEQ;
  const float scale1 = 2.0f / (float)SEQ;

  // per-lane row-select constants (branch-free A generation)
  const float k0 = (mA == 0) ? w0 : 0.0f;
  const float kc = (mA == 1) ? scale1 : 0.0f;
  const float ks = (mA == 2) ? scale1 : 0.0f;

  // initial angles for this lane's two K slots; rotation step = 4*theta
  float sn0, cs0, sn1, cs1, sD, cD;
  sincosf(theta * (float)(sbase + kh),     &sn0, &cs0);
  sincosf(theta * (float)(sbase + kh + 1), &sn1, &cs1);
  sincosf(4.0f * theta, &sD, &cD);

  v8f acc = {0.f, 0.f, 0.f, 0.f, 0.f, 0.f, 0.f, 0.f};

  // per-lane pointer into x[b, sbase+kh, h0+n]; advances 4 rows per iter
  const float* colp = x + ((size_t)b * SEQ + sbase + kh) * HID + h0 + n;

  #pragma unroll 4
  for (int t = 0; t < CHUNK_S / 4; ++t) {
    v2f bm;                       // B tile: rows kh, kh+1 of X[4,16]
    bm.x = colp[0];
    bm.y = colp[HID];
    colp += 4 * HID;

    v2f am;                       // A tile: weights for rows 0..2 (else 0)
    am.x = fmaf(kc, cs0, fmaf(ks, sn0, k0));
    am.y = fmaf(kc, cs1, fmaf(ks, sn1, k0));

    acc = __builtin_amdgcn_wmma_f32_16x16x4_f32(
        /*neg_a=*/false, am, /*neg_b=*/false, bm,
        /*c_mod=*/(short)0, acc, /*reuse_a=*/false, /*reuse_b=*/false);

    // rotate both angle pairs by 4*theta
    const float ncs0 = cs0 * cD - sn0 * sD;
    const float nsn0 = sn0 * cD + cs0 * sD;
    const float ncs1 = cs1 * cD - sn1 * sD;
    const float nsn1 = sn1 * cD + cs1 * sD;
    cs0 = ncs0; sn0 = nsn0; cs1 = ncs1; sn1 = nsn1;
  }

  // D rows 0..2 live in acc[0..2] on lanes 0..15
  if (lane < 16) {
    const int h = h0 + lane;
    const size_t base = ((size_t)b * SCHUNKS + chunk) * 3;
    partial[(base + 0) * HID + h] = acc[0];   // a0 partial
    partial[(base + 1) * HID + h] = acc[1];   // a1 partial
    partial[(base + 2) * HID + h] = acc[2];   // b1 partial
  }
}

// ---------------------------------------------------------------------------
// Pass 2: combine the SCHUNKS partials -> coef[b][j][h]
// ---------------------------------------------------------------------------
__global__ __launch_bounds__(256) void
freq_pass2_combine(const float* __restrict__ partial, float* __restrict__ coef) {
  const int idx = blockIdx.x * 256 + threadIdx.x;     // BATCH*3*HID total
  if (idx >= BATCH * 3 * HID) return;
  const int h = idx & (HID - 1);
  const int j = (idx >> 10) % 3;
  const int b = idx / (3 * HID);
  float s = 0.0f;
  #pragma unroll
  for (int c = 0; c < SCHUNKS; ++c)
    s += partial[(((size_t)b * SCHUNKS + c) * 3 + j) * HID + h];
  coef[idx] = s;
}

// ---------------------------------------------------------------------------
// Pass 3: fused low-pass reconstruction + residual + LayerNorm.
// One block = one (b, group of 32 s-rows). Coefficients loaded once/block;
// cos/sin advanced by a theta rotation recurrence (one sincosf per block).
// ---------------------------------------------------------------------------
#define ROWS_PER_BLOCK 32

__global__ __launch_bounds__(256) void
freq_pass3_ln(const float* __restrict__ x, const float* __restrict__ coef,
              const float* __restrict__ sqrt_beta, const float* __restrict__ gamma,
              const float* __restrict__ ln_beta, float* __restrict__ out) {
  __shared__ float2 red[2][8];

  const int b   = blockIdx.y;
  const int s0  = blockIdx.x * ROWS_PER_BLOCK;
  const int tid = threadIdx.x;
  const int h4  = tid * 4;
  const int lane = tid & 31;
  const int wv   = tid >> 5;

  // per-block constants (amortized over 32 rows)
  const float4 a0 = *(const float4*)(coef + ((size_t)b * 3 + 0) * HID + h4);
  const float4 a1 = *(const float4*)(coef + ((size_t)b * 3 + 1) * HID + h4);
  const float4 b1 = *(const float4*)(coef + ((size_t)b * 3 + 2) * HID + h4);
  const float4 sb = *(const float4*)(sqrt_beta + h4);
  const float4 g  = *(const float4*)(gamma + h4);
  const float4 be = *(const float4*)(ln_beta + h4);

  float4 cp, cm;   // (1+beta^2), (1-beta^2)
  {
    const float bx = sb.x * sb.x, by = sb.y * sb.y, bz = sb.z * sb.z, bw = sb.w * sb.w;
    cp.x = 1.f + bx; cp.y = 1.f + by; cp.z = 1.f + bz; cp.w = 1.f + bw;
    cm.x = 1.f - bx; cm.y = 1.f - by; cm.z = 1.f - bz; cm.w = 1.f - bw;
  }

  const float theta = TWO_PI / (float)SEQ;
  float sn, cs, sT, cT;
  sincosf(theta * (float)s0, &sn, &cs);   // angle for first row
  sincosf(theta, &sT, &cT);               // per-row rotation step

  const float* rowp = x + ((size_t)b * SEQ + s0) * HID + h4;
  float* outp       = out + ((size_t)b * SEQ + s0) * HID + h4;

  for (int r = 0; r < ROWS_PER_BLOCK; ++r) {
    const float4 xv = *(const float4*)rowp;
    if (r + 1 < ROWS_PER_BLOCK)
      __builtin_prefetch(rowp + HID, 0, 1);   // global_prefetch_b8: next row

    float4 hv;
    hv.x = cp.x * xv.x + cm.x * (a0.x + a1.x * cs + b1.x * sn);
    hv.y = cp.y * xv.y + cm.y * (a0.y + a1.y * cs + b1.y * sn);
    hv.z = cp.z * xv.z + cm.z * (a0.z + a1.z * cs + b1.z * sn);
    hv.w = cp.w * xv.w + cm.w * (a0.w + a1.w * cs + b1.w * sn);

    float sum = hv.x + hv.y + hv.z + hv.w;
    float ssq = hv.x * hv.x + hv.y * hv.y + hv.z * hv.z + hv.w * hv.w;

    // wave32 butterfly reduction
    #pragma unroll
    for (int off = 16; off > 0; off >>= 1) {
      sum += __shfl_xor(sum, off, 32);
      ssq += __shfl_xor(ssq, off, 32);
    }

    const int buf = r & 1;                 // double-buffered -> one barrier/row
    if (lane == 0) red[buf][wv] = make_float2(sum, ssq);
    __syncthreads();
    float tsum = 0.f, tssq = 0.f;
    #pragma unroll
    for (int w = 0; w < 8; ++w) { tsum += red[buf][w].x; tssq += red[buf][w].y; }

    const float mean = tsum * (1.0f / (float)HID);
    const float var  = tssq * (1.0f / (float)HID) - mean * mean;
    const float inv  = rsqrtf(var + 1e-12f);

    float4 o;
    o.x = (hv.x - mean) * inv * g.x + be.x;
    o.y = (hv.y - mean) * inv * g.y + be.y;
    o.z = (hv.z - mean) * inv * g.z + be.z;
    o.w = (hv.w - mean) * inv * g.w + be.w;
    *(float4*)outp = o;

    // rotate row angle by theta
    const float ncs = cs * cT - sn * sT;
    const float nsn = sn * cT + cs * sT;
    cs = ncs; sn = nsn;

    rowp += HID;
    outp += HID;
  }
}

// ---------------------------------------------------------------------------
extern "C" void kernel_launch(void* const* d_in, const int* in_sizes, int n_in,
                              void* d_out, int out_size, void* d_ws, size_t ws_size,
                              hipStream_t stream) {
  (void)in_sizes; (void)n_in; (void)out_size; (void)ws_size;
  const float* x         = (const float*)d_in[0];   // [32,2048,1024]
  const float* sqrt_beta = (const float*)d_in[1];   // [1,1,1024]
  const float* ln_gamma  = (const float*)d_in[2];   // [1024]
  const float* ln_beta   = (const float*)d_in[3];   // [1024]
  float* out = (float*)d_out;

  float* partial = (float*)d_ws;                                // 32*8*3*1024 f32 = 3 MB
  float* coef    = partial + (size_t)BATCH * SCHUNKS * 3 * HID; // 32*3*1024 f32 = 384 KB

  // Pass 1: 16384 waves (8 per block) -> chunked a0/a1/b1 partials via WMMA
  freq_pass1_wmma<<<BATCH * HTILES * SCHUNKS / 8, 256, 0, stream>>>(x, partial);

  // Pass 2: combine chunk partials
  freq_pass2_combine<<<(BATCH * 3 * HID + 255) / 256, 256, 0, stream>>>(partial, coef);

  // Pass 3: fused filter + residual + LayerNorm
  dim3 grid3(SEQ / ROWS_PER_BLOCK, BATCH);
  freq_pass3_ln<<<grid3, 256, 0, stream>>>(x, coef, sqrt_beta, ln_gamma, ln_beta, out);
}